// OT_Loss2_1417339208071
// MI455X (gfx1250) — compile-verified
//
#include <hip/hip_runtime.h>
#include <stdint.h>

// ---------------- problem constants (from reference) ----------------
#define WW      64                      // W = CROP/DS
#define NPIX    4096                    // W*W
#define NPTS    512                     // m
#define NB      16                      // batch
#define NITER   24                      // Michelot fixed-point iterations (exact simplex proj.)

#define GAMMA_F   10.0f
#define B_W       (1.0f / 512.0f)       // b = 1/m
#define BG_F      (GAMMA_F / 512.0f)    // b*gamma
#define INV_BG_F  (512.0f / GAMMA_F)    // 1/(b*gamma) = 51.2

static __device__ __forceinline__ float coodf(int i) { return (float)(i * 8 + 4); }

typedef float    v2f   __attribute__((ext_vector_type(2)));
typedef float    v8f   __attribute__((ext_vector_type(8)));
typedef uint32_t u32x4 __attribute__((ext_vector_type(4)));
typedef int      i32x4 __attribute__((ext_vector_type(4)));
typedef int      i32x8 __attribute__((ext_vector_type(8)));

// =====================================================================
// Main kernel: one block = 128 columns of one sample (8 waves x 16 cols)
//   Phase A: Michelot iteration -> exact simplex-projection threshold
//   Phase B: WMMA f32 16x16x4: D = A(alpha,ones) x B(G) accumulates
//            t_j = G^T alpha and sumG_j for 16 columns per wave.
// ws layout (floats): [0..63] per-block val partial, [64..127] per-block
// t partial, [128..143] aa per sample, [144..159] imgrad per sample.
// =====================================================================
__global__ __launch_bounds__(256) void k_cols(const float* __restrict__ pts,
                                              const float* __restrict__ alpha,
                                              float* __restrict__ ws)
{
    __shared__ float s_alpha[NPIX];     // 16 KB: alpha for this sample
    __shared__ float s_val[8], s_t[8];

    const int sample       = blockIdx.x >> 2;        // 4 blocks per sample
    const int colBlockBase = (blockIdx.x & 3) * 128;
    const int lane         = threadIdx.x & 31;
    const int wave         = threadIdx.x >> 5;

    // ---- TDM: stage alpha[sample][0..4095] into LDS (one DMA per block) ----
    if (threadIdx.x == 0) {
        uint64_t ga = (uint64_t)(uintptr_t)(alpha + (size_t)sample * NPIX);
        uint32_t la = (uint32_t)(uintptr_t)(&s_alpha[0]);      // LDS byte offset
        u32x4 g0 = { 1u,                                       // count=1 (valid user D#)
                     la,                                       // lds_addr
                     (uint32_t)(ga & 0xFFFFFFFFu),             // global_addr lo
                     (uint32_t)((ga >> 32) & 0x1FFFFFFu) | (2u << 30) }; // addr hi | type=2
        i32x8 g1 = { (int)(2u << 16),                          // data_size = 4 B
                     (int)((uint32_t)NPIX << 16),              // tensor_dim0 = 4096 (lo16)
                     (int)(1u << 16),                          // dim0 hi=0; tensor_dim1 = 1
                     (int)((uint32_t)NPIX << 16),              // dim1 hi=0; tile_dim0 = 4096
                     1,                                        // tile_dim1 = 1
                     NPIX,                                     // tensor_dim0_stride
                     0, 0 };
        i32x4 g2 = { 0, 0, 0, 0 };                             // (VADDR2: unused / NULL)
        i32x4 g3 = { 0, 0, 0, 0 };                             // (VADDR3: unused / NULL)
        i32x8 g4 = { 0, 0, 0, 0, 0, 0, 0, 0 };                 // extra group (clang-23 form)
        __builtin_amdgcn_tensor_load_to_lds(g0, g1, g2, g3, g4, 0);
        __builtin_amdgcn_s_wait_tensorcnt(0);
    }
    __syncthreads();

    // ---------------- Phase A: Michelot threshold, 2 lanes / column --------
    float theta;
    {
        const int   cA = colBlockBase + wave * 16 + (lane >> 1);
        const float xj = pts[((size_t)sample * NPTS + cA) * 2 + 0];
        const float yj = pts[((size_t)sample * NPTS + cA) * 2 + 1];
        const int   h  = lane & 1;                 // which half of the 4096 pixels
        float th = -1.0e30f;                       // first pass => theta0 = (sumV-1)/n
        for (int it = 0; it < NITER; ++it) {
            float S = 0.0f, K = 0.0f;
            for (int yi = h * 32; yi < h * 32 + 32; ++yi) {
                const float dy  = coodf(yi) - yj;
                const float dy2 = dy * dy;
                const float* ar = &s_alpha[yi * WW];
                for (int xi = 0; xi < WW; ++xi) {
                    const float dx = coodf(xi) - xj;
                    const float c  = fmaf(dx, dx, dy2);        // C(p,j) on the fly
                    const float v  = (ar[xi] - c) * INV_BG_F;  // V = (alpha-C)/(b*gamma)
                    if (v > th) { S += v; K += 1.0f; }
                }
            }
            S += __shfl_xor(S, 1, 32);             // combine the two half-column lanes
            K += __shfl_xor(K, 1, 32);
            th = (S - 1.0f) / fmaxf(K, 1.0f);      // theta <- (sum_active - 1)/k
        }
        theta = th;
    }

    // ---------------- Phase B: WMMA reduction over p, 16 columns/wave ------
    // A (16x4 f32): row0 = alpha[p..p+3], row1 = ones, rows 2..15 = 0.
    // B (4x16 f32): B[k][col] = G[p+k, col].  D row0 = t_j, row1 = sumG_j.
    // Lane mapping follows ISA A-matrix striping: lanes<16 hold K={0,1},
    // lanes>=16 hold K={2,3}; lane&15 = matrix column / A-row index.
    const int   col = lane & 15;
    const int   hi  = lane >> 4;
    const int   cB  = colBlockBase + wave * 16 + col;
    const float xj  = pts[((size_t)sample * NPTS + cB) * 2 + 0];
    const float yj  = pts[((size_t)sample * NPTS + cB) * 2 + 1];
    const float th  = __shfl(theta, col << 1, 32);   // theta lives on lane 2*col

    v8f   acc = { 0.f, 0.f, 0.f, 0.f, 0.f, 0.f, 0.f, 0.f };
    float g2p = 0.0f;
    for (int base = 0; base < NPIX; base += 4) {
        const int   p0 = base + (hi << 1);
        const int   p1 = p0 + 1;
        const float a0 = s_alpha[p0];
        const float a1 = s_alpha[p1];
        float G0, G1;
        {
            const float dx = coodf(p0 & 63) - xj;
            const float dy = coodf(p0 >> 6) - yj;
            const float v  = (a0 - fmaf(dx, dx, dy * dy)) * INV_BG_F;
            G0 = fmaxf(v - th, 0.0f);
        }
        {
            const float dx = coodf(p1 & 63) - xj;
            const float dy = coodf(p1 >> 6) - yj;
            const float v  = (a1 - fmaf(dx, dx, dy * dy)) * INV_BG_F;
            G1 = fmaxf(v - th, 0.0f);
        }
        g2p = fmaf(G0, G0, g2p);                    // sum G^2 on VALU (co-executes)
        g2p = fmaf(G1, G1, g2p);
        v2f Bv; Bv[0] = G0; Bv[1] = G1;
        v2f Av;
        Av[0] = (col == 0) ? a0 : ((col == 1) ? 1.0f : 0.0f);
        Av[1] = (col == 0) ? a1 : ((col == 1) ? 1.0f : 0.0f);
        acc = __builtin_amdgcn_wmma_f32_16x16x4_f32(false, Av, false, Bv,
                                                    (short)0, acc, false, false);
    }
    const float g2c  = g2p + __shfl_xor(g2p, 16, 32);   // K{0,1} + K{2,3} halves
    const float tcol = acc[0];                          // D[M=0][N=col] = sum alpha*G
    const float sumG = acc[1];                          // D[M=1][N=col] = sum G
    // val_j = sum X*G - 0.5*gamma*b*sum G^2 = 0.5*bg*g2 + bg*theta*sumG
    float v_c = fmaf(0.5f * BG_F, g2c, BG_F * th * sumG);
    float t_c = tcol;
    if (lane >= 16) { v_c = 0.0f; t_c = 0.0f; }
    #pragma unroll
    for (int o = 1; o < 16; o <<= 1) {
        v_c += __shfl_xor(v_c, o, 32);
        t_c += __shfl_xor(t_c, o, 32);
    }
    if (lane == 0) { s_val[wave] = v_c; s_t[wave] = t_c; }
    __syncthreads();
    if (threadIdx.x == 0) {                 // deterministic fixed-order block sum
        float vs = 0.f, ts = 0.f;
        for (int w = 0; w < 8; ++w) { vs += s_val[w]; ts += s_t[w]; }
        ws[blockIdx.x]      = vs;
        ws[64 + blockIdx.x] = ts;
    }
}

// =====================================================================
// Per-sample scalars: aa = <nd,alpha>, and the (near-zero) im_grad term.
// =====================================================================
__global__ __launch_bounds__(256) void k_sample(const float* __restrict__ nd,
                                                const float* __restrict__ ud,
                                                const float* __restrict__ alpha,
                                                float* __restrict__ ws)
{
    const int i = blockIdx.x;
    const float* ndp = nd    + (size_t)i * NPIX;
    const float* udp = ud    + (size_t)i * NPIX;
    const float* ap  = alpha + (size_t)i * NPIX;
    __builtin_prefetch(ndp + threadIdx.x, 0, 0);
    __builtin_prefetch(udp + threadIdx.x, 0, 0);

    float sc = 0.f, sa = 0.f, aa = 0.f;
    for (int p = threadIdx.x; p < NPIX; p += 256) {
        const float u = udp[p], al = ap[p];
        sc += u;
        sa  = fmaf(u, al, sa);
        aa  = fmaf(ndp[p], al, aa);
    }
    __shared__ float r0[8], r1[8], r2[8];
    __shared__ float tot[3];
    const int lane = threadIdx.x & 31, wave = threadIdx.x >> 5;
    #pragma unroll
    for (int o = 1; o < 32; o <<= 1) {
        sc += __shfl_xor(sc, o, 32);
        sa += __shfl_xor(sa, o, 32);
        aa += __shfl_xor(aa, o, 32);
    }
    if (lane == 0) { r0[wave] = sc; r1[wave] = sa; r2[wave] = aa; }
    __syncthreads();
    if (threadIdx.x == 0) {
        float a_ = 0.f, b_ = 0.f, c_ = 0.f;
        for (int w = 0; w < 8; ++w) { a_ += r0[w]; b_ += r1[w]; c_ += r2[w]; }
        tot[0] = a_; tot[1] = b_; tot[2] = c_;
    }
    __syncthreads();
    const float scT = tot[0], saT = tot[1], aaT = tot[2];
    const float den = fmaf(scT, scT, 1e-16f);
    const float c1  = scT / den;            // sc/(sc^2+eps)
    const float ig2 = saT / den;            // <sd,alpha>/(sc^2+eps)
    float img = 0.f;
    for (int p = threadIdx.x; p < NPIX; p += 256)
        img += udp[p] * fmaf(c1, ap[p], -ig2);
    #pragma unroll
    for (int o = 1; o < 32; o <<= 1) img += __shfl_xor(img, o, 32);
    __syncthreads();
    if (lane == 0) r0[wave] = img;
    __syncthreads();
    if (threadIdx.x == 0) {
        float s = 0.f;
        for (int w = 0; w < 8; ++w) s += r0[w];
        ws[128 + i] = aaT;
        ws[144 + i] = s;
    }
}

// =====================================================================
// Finalize: obj_i = aa - b*sum val ; loss_i = img + 0.1*((aa-b*sum t)-obj)
// =====================================================================
__global__ void k_final(const float* __restrict__ ws, float* __restrict__ out)
{
    const int l = threadIdx.x;
    float loss = 0.f, obj = 0.f;
    if (l < NB) {
        float val = 0.f, t = 0.f;
        for (int q = 0; q < 4; ++q) { val += ws[4 * l + q]; t += ws[64 + 4 * l + q]; }
        const float aa  = ws[128 + l];
        const float img = ws[144 + l];
        const float o   = aa - B_W * val;
        const float sad = aa - B_W * t;
        obj  = o;
        loss = img + 0.1f * (sad - o);
    }
    #pragma unroll
    for (int o2 = 1; o2 < 16; o2 <<= 1) {
        loss += __shfl_xor(loss, o2, 32);
        obj  += __shfl_xor(obj,  o2, 32);
    }
    if (l == 0) { out[0] = loss; out[1] = obj; }
}

extern "C" void kernel_launch(void* const* d_in, const int* in_sizes, int n_in,
                              void* d_out, int out_size, void* d_ws, size_t ws_size,
                              hipStream_t stream)
{
    (void)in_sizes; (void)n_in; (void)out_size; (void)ws_size;
    const float* nd    = (const float*)d_in[0];   // normed_density   [16,64,64]
    const float* ud    = (const float*)d_in[1];   // unnormed_density [16,64,64]
    const float* pts   = (const float*)d_in[2];   // points           [16,512,2]
    const float* alpha = (const float*)d_in[3];   // alpha_pred       [16,4096]
    float* ws  = (float*)d_ws;                    // 160 floats used
    float* out = (float*)d_out;                   // [loss, ot_obj]

    hipLaunchKernelGGL(k_sample, dim3(NB),     dim3(256), 0, stream, nd, ud, alpha, ws);
    hipLaunchKernelGGL(k_cols,   dim3(NB * 4), dim3(256), 0, stream, pts, alpha, ws);
    hipLaunchKernelGGL(k_final,  dim3(1),      dim3(32),  0, stream, ws, out);
}